// CustomPartiallyConnectedLayer_73813307949248
// MI455X (gfx1250) — compile-verified
//
#include <hip/hip_runtime.h>

// out[b, j] = sum_{c<128} x[b, j*128+c] * w[j*128+c] + bias[j]
// x: [2048, 65536] f32, w: [65536] f32, bias: [512] f32, out: [2048, 512] f32

#define H1 65536
#define H2 512
#define BATCH 2048

typedef float v2f __attribute__((ext_vector_type(2)));
typedef float v4f __attribute__((ext_vector_type(4)));
typedef float v8f __attribute__((ext_vector_type(8)));

// One wave computes 16 outputs: batch b, columns jbase..jbase+15.
// Per-lane FMA accumulators are laid out exactly as the WMMA f32 16x16x4
// A-matrix (lane L -> row M=L%16, its 2 VGPRs are 2 of that row's 4 K-slots).
// One v_wmma_f32_16x16x4_f32 against an all-ones B produces the 16 row sums.
__global__ __launch_bounds__(256) void pcl_wmma_kernel(
    const float* __restrict__ x,
    const float* __restrict__ w,
    const float* __restrict__ bias,
    float* __restrict__ out) {
  const int tid  = threadIdx.x;
  const int lane = tid & 31;
  const int wave = tid >> 5;          // 0..7
  const int m    = lane & 15;         // A-matrix row owned by this lane
  const int half = lane >> 4;         // which 64 of the 128 c's this lane sums

  const int b = blockIdx.x;           // one batch row per block
  const float* __restrict__ xrow = x + (size_t)b * H1;

  // Each wave handles 4 j-tiles of 16 columns => 8 waves cover all 512 j.
#pragma unroll 1
  for (int i = 0; i < 4; ++i) {
    const int jt = wave * 4 + i;                       // 0..31
    const int colbase = jt * 2048 + m * 128 + half * 64;

    float a0 = 0.0f, a1 = 0.0f;                        // A-matrix K-slots
#pragma unroll
    for (int t = 0; t < 16; ++t) {
      const int col = colbase + t * 4;
      v4f xv = *(const v4f*)(xrow + col);              // global_load_b128
      v4f wv = *(const v4f*)(w + col);                 // global_load_b128 (L2-hot)
      a0 = __builtin_fmaf(xv.x, wv.x, a0);
      a0 = __builtin_fmaf(xv.y, wv.y, a0);
      a1 = __builtin_fmaf(xv.z, wv.z, a1);
      a1 = __builtin_fmaf(xv.w, wv.w, a1);
    }

    v2f A; A.x = a0; A.y = a1;                         // 16x4 A tile (partials)
    v2f B; B.x = 1.0f; B.y = 1.0f;                     // 4x16 all-ones
    v8f C = {};
    // D[m][n] = sum_k A[m][k]  (identical across n): the 16 segment sums.
    v8f D = __builtin_amdgcn_wmma_f32_16x16x4_f32(
        /*neg_a=*/false, A, /*neg_b=*/false, B,
        /*c_mod=*/(short)0, C, /*reuse_a=*/false, /*reuse_b=*/false);

    // D VGPR r: lanes 0-15 hold (M=r, N=lane); lanes 16-31 hold (M=8+r).
    // Lane 0 -> rows 0..7, lane 16 -> rows 8..15 (column N=0 of each half).
    if (m == 0) {
      const int jbase = jt * 16 + half * 8;
      const float* __restrict__ bi = bias + jbase;
      float* __restrict__ o = out + (size_t)b * H2 + jbase;
      v4f o0, o1;
      o0.x = D[0] + bi[0];
      o0.y = D[1] + bi[1];
      o0.z = D[2] + bi[2];
      o0.w = D[3] + bi[3];
      o1.x = D[4] + bi[4];
      o1.y = D[5] + bi[5];
      o1.z = D[6] + bi[6];
      o1.w = D[7] + bi[7];
      *(v4f*)(o + 0) = o0;                             // global_store_b128
      *(v4f*)(o + 4) = o1;
    }
  }
}

extern "C" void kernel_launch(void* const* d_in, const int* in_sizes, int n_in,
                              void* d_out, int out_size, void* d_ws, size_t ws_size,
                              hipStream_t stream) {
  const float* x    = (const float*)d_in[0];
  const float* wts  = (const float*)d_in[1];
  const float* bias = (const float*)d_in[2];
  float* out        = (float*)d_out;

  dim3 grid(BATCH);   // 2048 blocks, 8 waves each -> 16384 waves
  dim3 block(256);
  pcl_wmma_kernel<<<grid, block, 0, stream>>>(x, wts, bias, out);
}